// SATUpsampler_77077483094148
// MI455X (gfx1250) — compile-verified
//
#include <hip/hip_runtime.h>
#include <math.h>

// ---------------- constants ----------------
#define Bn   4
#define Cn   64
#define Hn   128
#define Wn   128
#define HWn  (Hn*Wn)          // 16384
#define NP2n 16
#define KKn  25               // 5x5
#define HPn  512
#define WPn  512

typedef __attribute__((ext_vector_type(2))) float v2f;
typedef __attribute__((ext_vector_type(4))) float v4f;
typedef __attribute__((ext_vector_type(8))) float v8f;

__device__ __forceinline__ float gelu_exact(float x) {
    return 0.5f * x * (1.0f + erff(x * 0.70710678118654752f));
}

// ---------------------------------------------------------------------------
// Kernel 1: depthwise 3x3 conv (zero pad) + exact GELU.  (B,C,H,W) -> same
// ---------------------------------------------------------------------------
__global__ __launch_bounds__(256) void dw3x3_gelu(const float* __restrict__ in,
                                                  const float* __restrict__ w,
                                                  float* __restrict__ out) {
    int idx = blockIdx.x * 256 + threadIdx.x;          // B*C*H*W = 4,194,304
    int x = idx & (Wn - 1);
    int y = (idx >> 7) & (Hn - 1);
    int c = (idx >> 14) & (Cn - 1);
    int b = idx >> 20;
    const float* wp = w + c * 9;
    const float* ip = in + (b * Cn + c) * HWn;
    float acc = 0.0f;
#pragma unroll
    for (int ky = 0; ky < 3; ++ky) {
#pragma unroll
        for (int kx = 0; kx < 3; ++kx) {
            int yy = y + ky - 1, xx = x + kx - 1;
            if (yy >= 0 && yy < Hn && xx >= 0 && xx < Wn)
                acc += wp[ky * 3 + kx] * ip[yy * Wn + xx];
        }
    }
    out[idx] = gelu_exact(acc);
}

// ---------------------------------------------------------------------------
// Kernel 2: dense 3x3 conv C->C (zero pad) + GELU via implicit GEMM, fp32 WMMA.
// Block = 128 threads (4 waves). Block tile: 16 pixels x 64 out channels.
// Wave w computes out-channels [16w,16w+16). K loop: 9 taps x 16 ci-chunks.
// ---------------------------------------------------------------------------
__global__ __launch_bounds__(128) void conv3x3_wmma_gelu(const float* __restrict__ in,
                                                         const float* __restrict__ w,
                                                         float* __restrict__ out) {
    __shared__ float sIn[Cn][3][20];                   // 15360 B (cols 0..17 used)
    int blk = blockIdx.x;                              // b*1024 + y*8 + xt
    int xt = blk & 7;
    int y  = (blk >> 3) & (Hn - 1);
    int b  = blk >> 10;
    int x0 = xt * 16;

    for (int i = threadIdx.x; i < Cn * 3 * 18; i += 128) {
        int cc = i % 18;
        int r  = (i / 18) % 3;
        int ci = i / 54;
        int yy = y + r - 1, xx = x0 + cc - 1;
        float v = 0.0f;
        if (yy >= 0 && yy < Hn && xx >= 0 && xx < Wn)
            v = in[((b * Cn + ci) * Hn + yy) * Wn + xx];
        sIn[ci][r][cc] = v;
    }
    __syncthreads();

    int wv = threadIdx.x >> 5;
    int lane = threadIdx.x & 31;
    int m = lane & 15;                                 // M row (A) / N col (B)
    int koff = (lane >> 4) << 1;                       // K offset 0 or 2
    int co = wv * 16;

    v8f acc = {};
    for (int tap = 0; tap < 9; ++tap) {
        int ky = tap / 3, kx = tap % 3;
#pragma unroll
        for (int ci0 = 0; ci0 < Cn; ci0 += 4) {
            v2f a, bb;
            a.x  = w[((co + m) * Cn + ci0 + koff) * 9 + tap];
            a.y  = w[((co + m) * Cn + ci0 + koff + 1) * 9 + tap];
            bb.x = sIn[ci0 + koff][ky][kx + m];
            bb.y = sIn[ci0 + koff + 1][ky][kx + m];
            acc = __builtin_amdgcn_wmma_f32_16x16x4_f32(false, a, false, bb,
                                                        (short)0, acc, false, false);
        }
    }
    int rowadd = (lane >> 4) * 8;
#pragma unroll
    for (int r = 0; r < 8; ++r) {
        int oc = co + r + rowadd;
        out[((b * Cn + oc) * Hn + y) * Wn + x0 + m] = gelu_exact(acc[r]);
    }
}

// ---------------------------------------------------------------------------
// Kernel 3: 1x1 conv C=64 -> 400 (softmax logits), fp32 WMMA GEMM.
// Block = 128 threads (4 waves), 16-pixel N tile, waves sweep 25 M tiles.
// ---------------------------------------------------------------------------
__global__ __launch_bounds__(128) void conv1x1_400_wmma(const float* __restrict__ in,
                                                        const float* __restrict__ w,
                                                        float* __restrict__ out) {
    __shared__ float sF[Cn][16];                       // 4096 B
    int blk = blockIdx.x;                              // b*1024 + ptile
    int p = blk & 1023;
    int b = blk >> 10;
    int pix0 = p * 16;

    // stage 64x16 activation tile as b128 transfers
    for (int i = threadIdx.x; i < Cn * 4; i += 128) {
        int q = i & 3, ci = i >> 2;
        *(v4f*)&sF[ci][q * 4] =
            *(const v4f*)(in + (size_t)(b * Cn + ci) * HWn + pix0 + q * 4);
    }
    __syncthreads();

    int wv = threadIdx.x >> 5;
    int lane = threadIdx.x & 31;
    int m = lane & 15;
    int koff = (lane >> 4) << 1;
    int rowadd = (lane >> 4) * 8;

    for (int mt = wv; mt < 25; mt += 4) {
        v8f acc = {};
#pragma unroll
        for (int ci0 = 0; ci0 < Cn; ci0 += 4) {
            v2f a, bb;
            a.x  = w[(mt * 16 + m) * Cn + ci0 + koff];
            a.y  = w[(mt * 16 + m) * Cn + ci0 + koff + 1];
            bb.x = sF[ci0 + koff][m];
            bb.y = sF[ci0 + koff + 1][m];
            acc = __builtin_amdgcn_wmma_f32_16x16x4_f32(false, a, false, bb,
                                                        (short)0, acc, false, false);
        }
#pragma unroll
        for (int r = 0; r < 8; ++r) {
            int oc = mt * 16 + r + rowadd;
            out[(size_t)(b * 400 + oc) * HWn + pix0 + m] = acc[r];
        }
    }
}

// ---------------------------------------------------------------------------
// Kernel 4: softmax over the 25 kernel taps, in place. One thread per (b,s,pix).
// ---------------------------------------------------------------------------
__global__ __launch_bounds__(256) void softmax25(float* __restrict__ lg) {
    int idx = blockIdx.x * 256 + threadIdx.x;          // 4*16*16384 = 1,048,576
    int pix = idx & (HWn - 1);
    int s = (idx >> 14) & 15;
    int b = idx >> 18;
    float* p = lg + (size_t)(b * 400 + s * KKn) * HWn + pix;
    float mx = -3.4e38f;
#pragma unroll
    for (int k = 0; k < KKn; ++k) mx = fmaxf(mx, p[(size_t)k * HWn]);
    float sum = 0.0f;
    float e[KKn];
#pragma unroll
    for (int k = 0; k < KKn; ++k) { e[k] = __expf(p[(size_t)k * HWn] - mx); sum += e[k]; }
    float inv = 1.0f / sum;
#pragma unroll
    for (int k = 0; k < KKn; ++k) p[(size_t)k * HWn] = e[k] * inv;
}

// ---------------------------------------------------------------------------
// Kernel 5: per-pixel einsum bckhw,bskhw->bcshw (reflect-pad-2 patches) fused
// with pixel-shuffle(4) scatter into out_base (B,C,512,512).
// Block = 256 threads, 16-pixel row tile. LDS: f (400x16) + patch (64x5x21).
// Output rows of 4 (same s>>2) are contiguous -> b128 stores.
// ---------------------------------------------------------------------------
__global__ __launch_bounds__(256) void einsum_ps(const float* __restrict__ xin,
                                                 const float* __restrict__ f,
                                                 float* __restrict__ ob) {
    __shared__ float sF[400][16];                      // 25600 B
    __shared__ float sP[Cn][5][21];                    // 26880 B
    int blk = blockIdx.x;                              // b*1024 + y*8 + xt
    int xt = blk & 7;
    int y  = (blk >> 3) & (Hn - 1);
    int b  = blk >> 10;
    int x0 = xt * 16;

    // stage f logits: 400 channels x 16 px, b128 transfers (1600 v4f)
    for (int i = threadIdx.x; i < 400 * 4; i += 256) {
        int q = i & 3, ch = i >> 2;
        *(v4f*)&sF[ch][q * 4] =
            *(const v4f*)(f + (size_t)(b * 400 + ch) * HWn + y * Wn + x0 + q * 4);
    }
    // stage reflect-padded input patch region
    for (int i = threadIdx.x; i < Cn * 5 * 20; i += 256) {
        int cc = i % 20;
        int r  = (i / 20) % 5;
        int ci = i / 100;
        int yy = y + r - 2;  yy = yy < 0 ? -yy : (yy > Hn - 1 ? 2 * Hn - 2 - yy : yy);
        int xx = x0 + cc - 2; xx = xx < 0 ? -xx : (xx > Wn - 1 ? 2 * Wn - 2 - xx : xx);
        sP[ci][r][cc] = xin[((b * Cn + ci) * Hn + yy) * Wn + xx];
    }
    __syncthreads();

    int c = threadIdx.x & 63;
    int g = threadIdx.x >> 6;                          // 4 pixel groups
    for (int j = 0; j < 4; ++j) {
        int px = g * 4 + j;
        float acc[NP2n];
#pragma unroll
        for (int s = 0; s < NP2n; ++s) acc[s] = 0.0f;
        for (int ky = 0; ky < 5; ++ky) {
#pragma unroll
            for (int kx = 0; kx < 5; ++kx) {
                float pv = sP[c][ky][px + kx];
                int k = ky * 5 + kx;
#pragma unroll
                for (int s = 0; s < NP2n; ++s) acc[s] += pv * sF[s * KKn + k][px];
            }
        }
        int X = x0 + px;
        float* orow = ob + (((size_t)b * Cn + c) * HPn + y * 4) * WPn + X * 4;
#pragma unroll
        for (int r1 = 0; r1 < 4; ++r1) {               // Y2 = y*4+r1, X2 = X*4+r2
            v4f o = { acc[r1 * 4 + 0], acc[r1 * 4 + 1],
                      acc[r1 * 4 + 2], acc[r1 * 4 + 3] };
            *(v4f*)(orow + (size_t)r1 * WPn) = o;
        }
    }
}

// ---------------------------------------------------------------------------
// Kernel 6: 1x1 conv C=64 -> 32 + bias (vector field), fp32 WMMA.
// Block = 64 threads (2 waves), 16-pixel tile, wave = one 16-channel M tile.
// ---------------------------------------------------------------------------
__global__ __launch_bounds__(64) void conv1x1_32_wmma(const float* __restrict__ in,
                                                      const float* __restrict__ w,
                                                      const float* __restrict__ bias,
                                                      float* __restrict__ out) {
    __shared__ float sF[Cn][16];
    int blk = blockIdx.x;
    int p = blk & 1023;
    int b = blk >> 10;
    int pix0 = p * 16;

    for (int i = threadIdx.x; i < Cn * 4; i += 64) {
        int q = i & 3, ci = i >> 2;
        *(v4f*)&sF[ci][q * 4] =
            *(const v4f*)(in + (size_t)(b * Cn + ci) * HWn + pix0 + q * 4);
    }
    __syncthreads();

    int mt = threadIdx.x >> 5;                         // 0 or 1
    int lane = threadIdx.x & 31;
    int m = lane & 15;
    int koff = (lane >> 4) << 1;
    int rowadd = (lane >> 4) * 8;

    v8f acc = {};
#pragma unroll
    for (int ci0 = 0; ci0 < Cn; ci0 += 4) {
        v2f a, bb;
        a.x  = w[(mt * 16 + m) * Cn + ci0 + koff];
        a.y  = w[(mt * 16 + m) * Cn + ci0 + koff + 1];
        bb.x = sF[ci0 + koff][m];
        bb.y = sF[ci0 + koff + 1][m];
        acc = __builtin_amdgcn_wmma_f32_16x16x4_f32(false, a, false, bb,
                                                    (short)0, acc, false, false);
    }
#pragma unroll
    for (int r = 0; r < 8; ++r) {
        int oc = mt * 16 + r + rowadd;
        out[(b * 32 + oc) * HWn + pix0 + m] = acc[r] + bias[oc];
    }
}

// ---------------------------------------------------------------------------
// Kernel 7: Scharr gradients of out_base (reflect pad 1 @512) + advection.
// out = out_base + gamma * (vx*gx + vy*gy). One thread per output element.
// ---------------------------------------------------------------------------
__global__ __launch_bounds__(256) void advect_final(const float* __restrict__ ob,
                                                    const float* __restrict__ v,
                                                    const float* __restrict__ gamma,
                                                    float* __restrict__ out) {
    int idx = blockIdx.x * 256 + threadIdx.x;          // 4*64*512*512 = 67,108,864
    int X = idx & (WPn - 1);
    int Y = (idx >> 9) & (HPn - 1);
    int c = (idx >> 18) & (Cn - 1);
    int b = idx >> 24;

    const float* pb = ob + ((size_t)(b * Cn + c)) * HPn * WPn;
    int ym = Y - 1; ym = ym < 0 ? -ym : ym;
    int yp = Y + 1; yp = yp > HPn - 1 ? 2 * HPn - 2 - yp : yp;
    int xm = X - 1; xm = xm < 0 ? -xm : xm;
    int xp = X + 1; xp = xp > WPn - 1 ? 2 * WPn - 2 - xp : xp;

    float n00 = pb[(size_t)ym * WPn + xm], n01 = pb[(size_t)ym * WPn + X], n02 = pb[(size_t)ym * WPn + xp];
    float n10 = pb[(size_t)Y  * WPn + xm], n11 = pb[(size_t)Y  * WPn + X], n12 = pb[(size_t)Y  * WPn + xp];
    float n20 = pb[(size_t)yp * WPn + xm], n21 = pb[(size_t)yp * WPn + X], n22 = pb[(size_t)yp * WPn + xp];

    const float inv32 = 1.0f / 32.0f;
    float gx = (3.0f * (n02 - n00) + 10.0f * (n12 - n10) + 3.0f * (n22 - n20)) * inv32;
    float gy = (3.0f * (n20 - n00) + 10.0f * (n21 - n01) + 3.0f * (n22 - n02)) * inv32;

    int y = Y >> 2, x = X >> 2;
    int s = ((Y & 3) << 2) | (X & 3);
    float vx = v[((b * 32 + s) * Hn + y) * Wn + x];
    float vy = v[((b * 32 + 16 + s) * Hn + y) * Wn + x];

    out[idx] = n11 + gamma[0] * (vx * gx + vy * gy);
}

// ---------------------------------------------------------------------------
extern "C" void kernel_launch(void* const* d_in, const int* in_sizes, int n_in,
                              void* d_out, int out_size, void* d_ws, size_t ws_size,
                              hipStream_t stream) {
    const float* x     = (const float*)d_in[0];
    const float* fg_w1 = (const float*)d_in[1];
    const float* fg_w2 = (const float*)d_in[2];
    const float* fg_w3 = (const float*)d_in[3];
    const float* vg_w1 = (const float*)d_in[4];
    const float* vg_w2 = (const float*)d_in[5];
    const float* vg_b2 = (const float*)d_in[6];
    const float* gamma = (const float*)d_in[7];

    char* ws = (char*)d_ws;
    float* ob = (float*)ws;                               // 4*64*512*512*4 = 268,435,456 B
    float* lg = (float*)(ws + 268435456ull);              // 4*400*128*128*4 = 104,857,600 B
    float* t1 = (float*)(ws + 268435456ull + 104857600ull);        // 16 MB
    float* t2 = (float*)(ws + 268435456ull + 104857600ull + 16777216ull); // 16 MB

    // fg path
    dw3x3_gelu       <<<Bn * Cn * HWn / 256, 256, 0, stream>>>(x, fg_w1, t1);
    conv3x3_wmma_gelu<<<Bn * Hn * (Wn / 16), 128, 0, stream>>>(t1, fg_w2, t2);
    conv1x1_400_wmma <<<Bn * (HWn / 16), 128, 0, stream>>>(t2, fg_w3, lg);
    softmax25        <<<Bn * NP2n * HWn / 256, 256, 0, stream>>>(lg);
    einsum_ps        <<<Bn * Hn * (Wn / 16), 256, 0, stream>>>(x, lg, ob);

    // vg path (reuse t1/t2)
    dw3x3_gelu       <<<Bn * Cn * HWn / 256, 256, 0, stream>>>(x, vg_w1, t1);
    conv1x1_32_wmma  <<<Bn * (HWn / 16), 64, 0, stream>>>(t1, vg_w2, vg_b2, t2);

    // final advection
    advect_final     <<<Bn * Cn * HPn * WPn / 256, 256, 0, stream>>>(ob, t2, gamma, (float*)d_out);
}